// InvertAffine_10728828305755
// MI455X (gfx1250) — compile-verified
//
#include <hip/hip_runtime.h>

#define TPB 256
#define FLOATS_PER_SAMPLE 12   // 3x4 affine shift
#define BYTES_PER_SAMPLE  48

// Closed-form inverse of [[A+I, t],[0,1]] returned in shift form.
__device__ __forceinline__ void invert_affine12(const float* __restrict__ s,
                                                float* __restrict__ r) {
    const float a00 = s[0] + 1.0f, a01 = s[1],        a02 = s[2],        t0 = s[3];
    const float a10 = s[4],        a11 = s[5] + 1.0f, a12 = s[6],        t1 = s[7];
    const float a20 = s[8],        a21 = s[9],        a22 = s[10] + 1.0f, t2 = s[11];

    const float c00 = a11 * a22 - a12 * a21;
    const float c01 = a12 * a20 - a10 * a22;
    const float c02 = a10 * a21 - a11 * a20;
    const float det = a00 * c00 + a01 * c01 + a02 * c02;
    const float id  = 1.0f / det;

    const float r00 = c00 * id;
    const float r01 = (a02 * a21 - a01 * a22) * id;
    const float r02 = (a01 * a12 - a02 * a11) * id;
    const float r10 = c01 * id;
    const float r11 = (a00 * a22 - a02 * a20) * id;
    const float r12 = (a02 * a10 - a00 * a12) * id;
    const float r20 = c02 * id;
    const float r21 = (a01 * a20 - a00 * a21) * id;
    const float r22 = (a00 * a11 - a01 * a10) * id;

    r[0]  = r00 - 1.0f; r[1]  = r01;        r[2]  = r02;
    r[3]  = -(r00 * t0 + r01 * t1 + r02 * t2);
    r[4]  = r10;        r[5]  = r11 - 1.0f; r[6]  = r12;
    r[7]  = -(r10 * t0 + r11 * t1 + r12 * t2);
    r[8]  = r20;        r[9]  = r21;        r[10] = r22 - 1.0f;
    r[11] = -(r20 * t0 + r21 * t1 + r22 * t2);
}

__global__ __launch_bounds__(TPB) void InvertAffine_kernel(
    const float* __restrict__ in, float* __restrict__ out, int nSamples) {
    // Separate in/out staging tiles (12 KB each) -> one fewer barrier.
    // 24 KB / block still allows 13 workgroups per 320 KB WGP.
    __shared__ float ldsIn [TPB * FLOATS_PER_SAMPLE];
    __shared__ float ldsOut[TPB * FLOATS_PER_SAMPLE];

    const int tid       = threadIdx.x;
    const int blockBase = blockIdx.x * TPB;          // first sample of this block
    const bool fullTile = (blockBase + TPB) <= nSamples;

    if (fullTile) {
        const unsigned ldsInBase =
            (unsigned)(unsigned long long)(const void*)&ldsIn[0];   // LDS byte offset
        const unsigned ldsOutBase =
            (unsigned)(unsigned long long)(const void*)&ldsOut[0];

        // ---- Stage 1: async copy 12 KB global -> LDS, coalesced b128, non-temporal ----
        {
            const unsigned long long gbase =
                (unsigned long long)(const void*)in +
                (unsigned long long)blockBase * (unsigned long long)BYTES_PER_SAMPLE;
#pragma unroll
            for (int k = 0; k < 3; ++k) {
                const unsigned chunk = (unsigned)tid + (unsigned)k * TPB;  // 16B chunk id
                const unsigned lofs  = ldsInBase + chunk * 16u;
                const unsigned long long gaddr = gbase + (unsigned long long)chunk * 16ull;
                asm volatile("global_load_async_to_lds_b128 %0, %1, off th:TH_LOAD_NT"
                             :: "v"(lofs), "v"(gaddr)
                             : "memory");
            }
            asm volatile("s_wait_asynccnt 0x0" ::: "memory");  // my chunks are in LDS
        }
        __syncthreads();                                       // everyone's chunks visible

        // ---- Stage 2: per-thread closed-form inverse in registers ----
        float s[FLOATS_PER_SAMPLE], r[FLOATS_PER_SAMPLE];
#pragma unroll
        for (int j = 0; j < FLOATS_PER_SAMPLE; ++j)
            s[j] = ldsIn[tid * FLOATS_PER_SAMPLE + j];

        invert_affine12(s, r);

#pragma unroll
        for (int j = 0; j < FLOATS_PER_SAMPLE; ++j)
            ldsOut[tid * FLOATS_PER_SAMPLE + j] = r[j];
        __syncthreads();                                       // results staged

        // ---- Stage 3: async store LDS -> global, coalesced b128, non-temporal ----
        {
            const unsigned long long obase =
                (unsigned long long)(void*)out +
                (unsigned long long)blockBase * (unsigned long long)BYTES_PER_SAMPLE;
#pragma unroll
            for (int k = 0; k < 3; ++k) {
                const unsigned chunk = (unsigned)tid + (unsigned)k * TPB;
                const unsigned lofs  = ldsOutBase + chunk * 16u;
                const unsigned long long gaddr = obase + (unsigned long long)chunk * 16ull;
                asm volatile("global_store_async_from_lds_b128 %0, %1, off th:TH_STORE_NT"
                             :: "v"(gaddr), "v"(lofs)
                             : "memory");
            }
            asm volatile("s_wait_asynccnt 0x0" ::: "memory");  // flush before endpgm
        }
    } else {
        // Tail block: guarded scalar path (branch is block-uniform, no barriers here)
        const int sIdx = blockBase + tid;
        if (sIdx < nSamples) {
            float s[FLOATS_PER_SAMPLE], r[FLOATS_PER_SAMPLE];
            const float* p = in + (size_t)sIdx * FLOATS_PER_SAMPLE;
#pragma unroll
            for (int j = 0; j < FLOATS_PER_SAMPLE; ++j) s[j] = p[j];
            invert_affine12(s, r);
            float* q = out + (size_t)sIdx * FLOATS_PER_SAMPLE;
#pragma unroll
            for (int j = 0; j < FLOATS_PER_SAMPLE; ++j) q[j] = r[j];
        }
    }
}

extern "C" void kernel_launch(void* const* d_in, const int* in_sizes, int n_in,
                              void* d_out, int out_size, void* d_ws, size_t ws_size,
                              hipStream_t stream) {
    (void)n_in; (void)out_size; (void)d_ws; (void)ws_size;
    const float* trf = (const float*)d_in[0];
    float* out       = (float*)d_out;
    const int nSamples = in_sizes[0] / FLOATS_PER_SAMPLE;  // 4,194,304
    const int blocks   = (nSamples + TPB - 1) / TPB;
    InvertAffine_kernel<<<blocks, TPB, 0, stream>>>(trf, out, nSamples);
}